// MessagePassingLayer_85564338471237
// MI455X (gfx1250) — compile-verified
//
#include <hip/hip_runtime.h>
#include <hip/hip_bf16.h>

typedef float v2f  __attribute__((ext_vector_type(2)));
typedef float v8f  __attribute__((ext_vector_type(8)));
typedef unsigned int u32x4 __attribute__((ext_vector_type(4)));
typedef int          i32x4 __attribute__((ext_vector_type(4)));
typedef int          i32x8 __attribute__((ext_vector_type(8)));

// ---------------- helpers ----------------
__device__ __forceinline__ unsigned fenc(float f) {
    unsigned u = __float_as_uint(f);
    return (u & 0x80000000u) ? ~u : (u | 0x80000000u);   // order-preserving float->uint
}
__device__ __forceinline__ float fdec(unsigned k) {
    unsigned u = (k & 0x80000000u) ? (k ^ 0x80000000u) : ~k;
    return __uint_as_float(u);
}
__device__ __forceinline__ float leaky(float x) { return (x >= 0.f) ? x : 0.01f * x; }

#define ENC_NEG_INF 0x007FFFFFu   // encoded(-inf)

// W-tile staging geometry: K-chunk of 32 dwords per row, TDM pads +4 dwords per 32
constexpr int KCHUNK = 32;
constexpr int WROW   = 36;          // 32 + 4 pad -> LDS row pitch (bank-conflict-free)
constexpr int WBUF   = 128 * WROW;  // floats per double-buffer half

// ---- TDM: DMA a [KCHUNK x rows] 2D fp32 tile (row-major, row length = rowlen) into LDS ----
// D# per CDNA5 ISA ch.8: group0 = {count|flags, lds_addr, global_addr, type=2},
// group1 = {data_size=4B, pad(32dw -> +4dw), tensor dims, tile dims, dim0 stride}.
__device__ __forceinline__ void tdm_load_w_tile(const float* gsrc, unsigned lds_byte_off,
                                                unsigned rowlen, unsigned rows) {
    unsigned long long ga = (unsigned long long)(size_t)gsrc;
    u32x4 g0;
    g0.x = 1u;                                                  // count=1, user mode, no gather
    g0.y = lds_byte_off;                                        // lds_addr (bytes)
    g0.z = (unsigned)ga;                                        // global_addr[31:0]
    g0.w = (unsigned)((ga >> 32) & 0x01FFFFFFull) | (2u << 30); // global_addr[56:32] | type=2
    i32x8 g1;
    g1[0] = (int)((2u << 16)            // data_size = 4 bytes
                | (1u << 20)            // pad_enable
                | (4u << 22)            // pad_interval code 4 -> every 32 dwords
                | (3u << 25));          // pad_amount  code 3 -> 4 dwords
    g1[1] = (int)((rowlen & 0xFFFFu) << 16);                    // tensor_dim0[15:0]
    g1[2] = (int)((rowlen >> 16) | ((rows & 0xFFFFu) << 16));   // dim0[31:16] | dim1[15:0]
    g1[3] = (int)((rows >> 16) | ((unsigned)KCHUNK << 16));     // dim1[31:16] | tile_dim0
    g1[4] = (int)(rows & 0xFFFFu);                              // tile_dim1 (tile_dim2 = 0)
    g1[5] = (int)rowlen;                                        // tensor_dim0_stride[31:0]
    g1[6] = 0;                                                  // stride[47:32] | dim1_stride lo
    g1[7] = 0;
    i32x4 z4 = {0, 0, 0, 0};
#if __clang_major__ >= 23
    i32x8 z8 = {0, 0, 0, 0, 0, 0, 0, 0};
    __builtin_amdgcn_tensor_load_to_lds(g0, g1, z4, z4, z8, 0);
#else
    __builtin_amdgcn_tensor_load_to_lds(g0, g1, z4, z4, 0);
#endif
}

// ---------------- kernel 0: init workspace ----------------
__global__ __launch_bounds__(256) void init_ws_kernel(unsigned* __restrict__ m_enc,
                                                      float* __restrict__ denom,
                                                      float* __restrict__ attn_feat, int N) {
    int i = blockIdx.x * 256 + threadIdx.x;
    if (i < N) { m_enc[i] = ENC_NEG_INF; denom[i] = 0.f; }
    if (i < N * 128) attn_feat[i] = 0.f;
}

// ---------------- kernel 1: edge update (gather->norm->WMMA GEMM->leaky->attn) ----------------
// Block: 256 threads (8 waves). Tile: 16 edges x 128 outputs. K = 384 in 12 TDM chunks of 32.
__global__ __launch_bounds__(256) void edge_update_kernel(
    const float* __restrict__ nfeat, const float* __restrict__ efeat,
    const float* __restrict__ W_e, const float* __restrict__ b_e, const float* __restrict__ a_e,
    const int* __restrict__ src, const int* __restrict__ dst,
    float* __restrict__ e_new, float* __restrict__ attn, unsigned* __restrict__ m_enc, int E) {
    constexpr int XP = 388;          // 384 + pad 4 (bank-conflict-free A-frag reads)
    constexpr int NCHUNK = 384 / KCHUNK;
    __shared__ float Xs[16 * XP];
    __shared__ float Ws[2][WBUF];    // double-buffered TDM destination
    __shared__ float attn_s[16];

    const int t  = threadIdx.x;
    const int e0 = blockIdx.x * 16;
    if (t < 16) attn_s[t] = 0.f;

    // ---- gather-concat + instance norm into LDS: row = edge, 384 cols ----
    {
        const int row = t >> 4, w = t & 15;
        const int e = min(e0 + row, E - 1);
        const int s = src[e], d = dst[e];
        float vals[24];
        float sum = 0.f, sq = 0.f;
#pragma unroll
        for (int j = 0; j < 24; ++j) {
            const int c = j * 16 + w;                 // lanes sweep consecutive columns
            float v;
            if (c < 128)       v = nfeat[(size_t)s * 128 + c];
            else if (c < 256)  v = efeat[(size_t)e * 128 + (c - 128)];
            else               v = nfeat[(size_t)d * 128 + (c - 256)];
            vals[j] = v; sum += v; sq += v * v;
        }
#pragma unroll
        for (int off = 8; off >= 1; off >>= 1) {      // reduce over the 16 workers of this row
            sum += __shfl_xor(sum, off, 16);
            sq  += __shfl_xor(sq,  off, 16);
        }
        const float mean = sum * (1.0f / 384.0f);
        const float var  = sq * (1.0f / 384.0f) - mean * mean;
        const float inv  = rsqrtf(var + 1e-5f);
#pragma unroll
        for (int j = 0; j < 24; ++j) Xs[row * XP + j * 16 + w] = (vals[j] - mean) * inv;
    }

    const int lane = t & 31, wave = t >> 5;
    const int half = lane >> 4, l16 = lane & 15;
    const int n = wave * 16 + l16;
    v8f acc = {0.f, 0.f, 0.f, 0.f, 0.f, 0.f, 0.f, 0.f};

    const unsigned ws_base = (unsigned)(size_t)(void*)&Ws[0][0];

    // prologue: wave 0 kicks chunk 0 DMA while the norm data settles
    if (t < 32) tdm_load_w_tile(W_e, ws_base, 384u, 128u);
    __syncthreads();   // Xs ready for all waves

    for (int c = 0; c < NCHUNK; ++c) {
        if (t < 32) __builtin_amdgcn_s_wait_tensorcnt(0);  // chunk c landed in LDS
        __syncthreads();
        if (t < 32 && (c + 1) < NCHUNK)                    // overlap next DMA with compute
            tdm_load_w_tile(W_e + (size_t)(c + 1) * KCHUNK,
                            ws_base + (unsigned)(((c + 1) & 1) * WBUF * 4), 384u, 128u);
        const float* wb = &Ws[c & 1][0];
        const int kc = c * KCHUNK;
#pragma unroll
        for (int k4 = 0; k4 < KCHUNK; k4 += 4) {
            const int kb = k4 + 2 * half;                  // ISA frag layout: K = vgpr + 2*half
            v2f a, b;
            a = *(const v2f*)(Xs + l16 * XP + kc + kb);
            b = *(const v2f*)(wb + n * WROW + kb);         // padded pitch 36 -> no bank conflict
            acc = __builtin_amdgcn_wmma_f32_16x16x4_f32(false, a, false, b, (short)0, acc,
                                                        false, false);
        }
        __syncthreads();   // chunk c consumed; its buffer may be overwritten by chunk c+2
    }

    // ---- epilogue: bias + leaky relu, store e_new, attention dot, segment max ----
    const float be = b_e[n], ae = a_e[n];
#pragma unroll
    for (int v = 0; v < 8; ++v) {
        const int m = v + 8 * half;                        // ISA C/D layout
        float x = leaky(acc[v] + be);
        if (e0 + m < E) e_new[(size_t)(e0 + m) * 128 + n] = x;
        float p = x * ae;
#pragma unroll
        for (int off = 8; off >= 1; off >>= 1) p += __shfl_xor(p, off, 16);
        if (l16 == 0) atomicAdd(&attn_s[m], p);            // ds_add_f32 across 8 waves
    }
    __syncthreads();
    if (t < 16 && e0 + t < E) {
        const float av = attn_s[t];
        attn[e0 + t] = av;
        atomicMax(&m_enc[dst[e0 + t]], fenc(av));          // segment max via encoded uint
    }
}

// ---------------- kernel 2: ex = exp(attn - m[dst]); denom[dst] += ex ----------------
__global__ __launch_bounds__(256) void softmax_denom_kernel(float* __restrict__ attn_ex,
                                                            const int* __restrict__ dst,
                                                            const unsigned* __restrict__ m_enc,
                                                            float* __restrict__ denom, int E) {
    const int e = blockIdx.x * 256 + threadIdx.x;
    if (e >= E) return;
    const int d = dst[e];
    const float m = fdec(m_enc[d]);
    const float v = expf(attn_ex[e] - m);
    attn_ex[e] = v;                                        // in-place: attn -> ex
    atomicAdd(&denom[d], v);
}

// ---------------- kernel 3: attn_feat[dst] += (ex/denom[dst]) * e_new ----------------
__global__ __launch_bounds__(256) void scatter_feat_kernel(const float* __restrict__ e_new,
                                                           const float* __restrict__ ex,
                                                           const float* __restrict__ denom,
                                                           const int* __restrict__ dst,
                                                           float* __restrict__ attn_feat, int E) {
    const int tid  = blockIdx.x * 256 + threadIdx.x;
    const int e    = tid >> 5;
    const int lane = tid & 31;
    if (e >= E) return;
    const int d = dst[e];
    const float alpha = ex[e] / denom[d];
    const int c = lane * 4;
    const float4 v = *(const float4*)(e_new + (size_t)e * 128 + c);
    float* o = attn_feat + (size_t)d * 128 + c;
    atomicAdd(o + 0, alpha * v.x);
    atomicAdd(o + 1, alpha * v.y);
    atomicAdd(o + 2, alpha * v.z);
    atomicAdd(o + 3, alpha * v.w);
}

// ---------------- kernel 4: node update (concat->norm->WMMA GEMM->leaky) ----------------
__global__ __launch_bounds__(256) void node_update_kernel(
    const float* __restrict__ nfeat, const float* __restrict__ attn_feat,
    const float* __restrict__ W_n, const float* __restrict__ b_n,
    float* __restrict__ out, int N) {
    constexpr int XP = 260;          // 256 + pad 4
    constexpr int NCHUNK = 256 / KCHUNK;
    __shared__ float Xs[16 * XP];
    __shared__ float Ws[2][WBUF];

    const int t  = threadIdx.x;
    const int n0 = blockIdx.x * 16;

    {
        const int row = t >> 4, w = t & 15;
        const int node = min(n0 + row, N - 1);
        float vals[16];
        float sum = 0.f, sq = 0.f;
#pragma unroll
        for (int j = 0; j < 16; ++j) {
            const int c = j * 16 + w;
            const float v = (c < 128) ? nfeat[(size_t)node * 128 + c]
                                      : attn_feat[(size_t)node * 128 + (c - 128)];
            vals[j] = v; sum += v; sq += v * v;
        }
#pragma unroll
        for (int off = 8; off >= 1; off >>= 1) {
            sum += __shfl_xor(sum, off, 16);
            sq  += __shfl_xor(sq,  off, 16);
        }
        const float mean = sum * (1.0f / 256.0f);
        const float var  = sq * (1.0f / 256.0f) - mean * mean;
        const float inv  = rsqrtf(var + 1e-5f);
#pragma unroll
        for (int j = 0; j < 16; ++j) Xs[row * XP + j * 16 + w] = (vals[j] - mean) * inv;
    }

    const int lane = t & 31, wave = t >> 5;
    const int half = lane >> 4, l16 = lane & 15;
    const int n = wave * 16 + l16;
    v8f acc = {0.f, 0.f, 0.f, 0.f, 0.f, 0.f, 0.f, 0.f};

    const unsigned ws_base = (unsigned)(size_t)(void*)&Ws[0][0];

    if (t < 32) tdm_load_w_tile(W_n, ws_base, 256u, 128u);
    __syncthreads();

    for (int c = 0; c < NCHUNK; ++c) {
        if (t < 32) __builtin_amdgcn_s_wait_tensorcnt(0);
        __syncthreads();
        if (t < 32 && (c + 1) < NCHUNK)
            tdm_load_w_tile(W_n + (size_t)(c + 1) * KCHUNK,
                            ws_base + (unsigned)(((c + 1) & 1) * WBUF * 4), 256u, 128u);
        const float* wb = &Ws[c & 1][0];
        const int kc = c * KCHUNK;
#pragma unroll
        for (int k4 = 0; k4 < KCHUNK; k4 += 4) {
            const int kb = k4 + 2 * half;
            v2f a, b;
            a = *(const v2f*)(Xs + l16 * XP + kc + kb);
            b = *(const v2f*)(wb + n * WROW + kb);
            acc = __builtin_amdgcn_wmma_f32_16x16x4_f32(false, a, false, b, (short)0, acc,
                                                        false, false);
        }
        __syncthreads();
    }

    const float bn = b_n[n];
#pragma unroll
    for (int v = 0; v < 8; ++v) {
        const int m = v + 8 * half;
        const float x = leaky(acc[v] + bn);
        if (n0 + m < N) out[(size_t)(n0 + m) * 128 + n] = x;
    }
}

// ---------------- host launcher ----------------
extern "C" void kernel_launch(void* const* d_in, const int* in_sizes, int n_in,
                              void* d_out, int out_size, void* d_ws, size_t ws_size,
                              hipStream_t stream) {
    (void)n_in; (void)out_size; (void)ws_size;
    const float* nfeat = (const float*)d_in[0];
    const float* efeat = (const float*)d_in[1];
    const float* W_e   = (const float*)d_in[2];
    const float* b_e   = (const float*)d_in[3];
    const float* a_e   = (const float*)d_in[4];
    const float* W_n   = (const float*)d_in[5];
    const float* b_n   = (const float*)d_in[6];
    const int*   src   = (const int*)d_in[7];
    const int*   dst   = (const int*)d_in[8];
    float* out = (float*)d_out;

    const int N = in_sizes[0] / 128;   // 50000
    const int E = in_sizes[7];         // 500000

    // workspace layout (~284 MB): e_new | attn/ex | denom | m_enc | attn_feat
    char* p = (char*)d_ws;
    float*    e_new     = (float*)p;    p += (size_t)E * 128 * sizeof(float);
    float*    attn_ex   = (float*)p;    p += (size_t)E * sizeof(float);
    float*    denom     = (float*)p;    p += (size_t)N * sizeof(float);
    unsigned* m_enc     = (unsigned*)p; p += (size_t)N * sizeof(unsigned);
    float*    attn_feat = (float*)p;

    init_ws_kernel<<<(N * 128 + 255) / 256, 256, 0, stream>>>(m_enc, denom, attn_feat, N);
    edge_update_kernel<<<(E + 15) / 16, 256, 0, stream>>>(nfeat, efeat, W_e, b_e, a_e, src, dst,
                                                          e_new, attn_ex, m_enc, E);
    softmax_denom_kernel<<<(E + 255) / 256, 256, 0, stream>>>(attn_ex, dst, m_enc, denom, E);
    scatter_feat_kernel<<<(E * 32 + 255) / 256, 256, 0, stream>>>(e_new, attn_ex, denom, dst,
                                                                  attn_feat, E);
    node_update_kernel<<<(N + 15) / 16, 256, 0, stream>>>(nfeat, attn_feat, W_n, b_n, out, N);
}